// LSTMModuleNetwork_60541859004898
// MI455X (gfx1250) — compile-verified
//
#include <hip/hip_runtime.h>
#include <math.h>

typedef __attribute__((ext_vector_type(2))) float v2f;
typedef __attribute__((ext_vector_type(8))) float v8f;

#define HID   768
#define I1    513          // INPUT_SIZE + 1
#define H4    3072         // 4*HID
#define NARMS 32
#define NU_C  0.1f

// flat param offsets (torch order: W_ih, W_hh, b_ih, b_hh, fc_W, fc_b)
#define N_IH    1575936u   // 3072*513
#define N_HH    2359296u   // 3072*768
#define BIH_OFF 3935232u   // N_IH+N_HH
#define BHH_OFF 3938304u
#define FCW_OFF 3941376u
#define FCB_OFF 3965952u
#define TOTALP  3965984u

// output layout: [selected_reward(1) | arm_probs(32) | selected_arm(1) | h_new(768) | c_new(768) | U_new(TOTALP)]
#define O_PROBS 1
#define O_ARM   33
#define O_H     34
#define O_C     802
#define O_U     1570

__device__ __forceinline__ float wave_sum(float v) {
    #pragma unroll
    for (int m = 16; m; m >>= 1) v += __shfl_xor(v, m, 32);
    return v;
}

// ---------- 1: build xin = [x; prev_reward], xin^2, h0^2 ----------
__global__ void prep_kernel(const float* __restrict__ x, const float* __restrict__ prev,
                            const float* __restrict__ h0,
                            float* __restrict__ xin, float* __restrict__ xin2,
                            float* __restrict__ h2) {
    int t = threadIdx.x;
    if (t < 512) { float v = x[t]; xin[t] = v; xin2[t] = v * v; }
    if (t == 512) { float v = prev[0]; xin[512] = v; xin2[512] = v * v; }
    if (t < HID) { float v = h0[t]; h2[t] = v * v; }
}

// ---------- 2: gates = W_ih@xin + W_hh@h0 + b_ih + b_hh via V_WMMA_F32_16X16X4_F32 ----------
// one wave (32 threads) per 16 gate rows. B has all 16 columns equal to the vector
// segment so every column of D holds the matvec partial -> layout-robust extraction.
// K loops are exact multiples of 4 (512 and 768): no guards, no divergence in the
// hot loop; the odd 513th input column is a rank-1 correction applied at writeout.
__global__ void gates_wmma_kernel(const float* __restrict__ Wih, const float* __restrict__ Whh,
                                  const float* __restrict__ bih, const float* __restrict__ bhh,
                                  const float* __restrict__ h0,  const float* __restrict__ xin,
                                  float* __restrict__ gates) {
    const int r0   = blockIdx.x * 16;          // gate-row base (192 blocks)
    const int lane = threadIdx.x & 31;
    const int m    = lane & 15;                // row within tile
    const int kh   = (lane >> 4) << 1;         // K half: lanes 0-15 -> k0,k1 ; 16-31 -> k2,k3
    v8f acc = {0.f,0.f,0.f,0.f,0.f,0.f,0.f,0.f};

    const float* Wrow = Wih + (size_t)(r0 + m) * I1;
    for (int k0 = 0; k0 < 512; k0 += 4) {      // main part of K=513, unguarded
        int ka = k0 + kh;
        v2f A = {Wrow[ka], Wrow[ka + 1]};
        v2f B = {xin[ka],  xin[ka + 1]};
        acc = __builtin_amdgcn_wmma_f32_16x16x4_f32(false, A, false, B, (short)0, acc, false, false);
    }
    const float* Hrow = Whh + (size_t)(r0 + m) * HID;
    for (int k0 = 0; k0 < HID; k0 += 4) {      // HID % 4 == 0
        int ka = k0 + kh;
        v2f A = {Hrow[ka], Hrow[ka + 1]};
        v2f B = {h0[ka],   h0[ka + 1]};
        acc = __builtin_amdgcn_wmma_f32_16x16x4_f32(false, A, false, B, (short)0, acc, false, false);
    }
    // C/D layout: VGPR n -> row n (lanes 0-15) and row n+8 (lanes 16-31); all cols equal.
    const float xlast = xin[512];
    if (lane == 0) {
        #pragma unroll
        for (int n = 0; n < 8; ++n) {
            int r = r0 + n;
            gates[r] = acc[n] + bih[r] + bhh[r] + Wih[(size_t)r * I1 + 512] * xlast;
        }
    } else if (lane == 16) {
        #pragma unroll
        for (int n = 0; n < 8; ++n) {
            int r = r0 + 8 + n;
            gates[r] = acc[n] + bih[r] + bhh[r] + Wih[(size_t)r * I1 + 512] * xlast;
        }
    }
}

// ---------- 3: activations, h_new/c_new, per-gate derivatives dg, h_new^2 ----------
__global__ void act_kernel(const float* __restrict__ gates, const float* __restrict__ c0,
                           float* __restrict__ out, float* __restrict__ dg,
                           float* __restrict__ hn2) {
    int k = blockIdx.x * blockDim.x + threadIdx.x;
    if (k >= HID) return;
    float gi = gates[k], gf = gates[HID + k], gg = gates[2 * HID + k], go = gates[3 * HID + k];
    float si = 1.f / (1.f + __expf(-gi));
    float sf = 1.f / (1.f + __expf(-gf));
    float so = 1.f / (1.f + __expf(-go));
    float tg = tanhf(gg);
    float c  = c0[k];
    float cn = sf * c + si * tg;
    float tc = tanhf(cn);
    float hn = so * tc;
    out[O_H + k] = hn;
    out[O_C + k] = cn;
    hn2[k] = hn * hn;
    float dtc = so * (1.f - tc * tc);
    dg[k]           = dtc * tg * si * (1.f - si);   // d h_new / d gate_i
    dg[HID + k]     = dtc * c  * sf * (1.f - sf);   // d h_new / d gate_f
    dg[2 * HID + k] = dtc * si * (1.f - tg * tg);   // d h_new / d gate_g
    dg[3 * HID + k] = tc * so * (1.f - so);         // d h_new / d gate_o
}

// ---------- 4: Wr[r] = sum_j xin2/U_Wih + sum_j h2/U_Whh + 1/U_bih + 1/U_bhh ----------
__global__ void wr_kernel(const float* __restrict__ U, const float* __restrict__ xin2,
                          const float* __restrict__ h2, float* __restrict__ Wr) {
    int r    = blockIdx.x * (blockDim.x >> 5) + (threadIdx.x >> 5); // 384 blocks * 8 waves = 3072
    int lane = threadIdx.x & 31;
    float s = 0.f;
    const float* Ui = U + (size_t)r * I1;
    for (int j = lane; j < I1; j += 32) s += xin2[j] / Ui[j];
    const float* Uh = U + N_IH + (size_t)r * HID;
    for (int j = lane; j < HID; j += 32) s += h2[j] / Uh[j];
    s = wave_sum(s);
    if (lane == 0) Wr[r] = s + 1.f / U[BIH_OFF + r] + 1.f / U[BHH_OFF + r];
}

// ---------- 5: per-arm rewards, T_fc, sigma2_lstm (one wave per arm) ----------
__global__ void arm_kernel(const float* __restrict__ fcW, const float* __restrict__ fcb,
                           const float* __restrict__ U,   const float* __restrict__ out,
                           const float* __restrict__ dg,  const float* __restrict__ Wr,
                           const float* __restrict__ hn2,
                           float* __restrict__ rew, float* __restrict__ Tfc,
                           float* __restrict__ s2l) {
    int a    = blockIdx.x;
    int lane = threadIdx.x & 31;
    float rw = 0.f, t = 0.f, s2 = 0.f;
    for (int j = lane; j < HID; j += 32) {
        float fw = fcW[a * HID + j];
        float hn = out[O_H + j];
        rw += fw * hn;
        t  += hn2[j] / U[FCW_OFF + a * HID + j];
        float e = 0.f;
        #pragma unroll
        for (int g = 0; g < 4; ++g) { float d = dg[g * HID + j]; e += d * d * Wr[g * HID + j]; }
        s2 += fw * fw * e;
    }
    rw = wave_sum(rw); t = wave_sum(t); s2 = wave_sum(s2);
    if (lane == 0) { rew[a] = rw + fcb[a]; Tfc[a] = t; s2l[a] = s2; }
}

// ---------- 6: UCB select + softmax (single wave, lane == arm) ----------
__global__ void select_kernel(const float* __restrict__ rew, const float* __restrict__ Tfc,
                              const float* __restrict__ s2l, const float* __restrict__ U,
                              float* __restrict__ out, int* __restrict__ sel) {
    int a = threadIdx.x;
    float pre = 0.f;
    for (int ap = 0; ap <= a; ++ap) pre += Tfc[ap] + 1.f / U[FCB_OFF + ap];  // cumsum from fc grads
    float r = rew[a];
    float score = r + NU_C * sqrtf(s2l[a] + pre);
    // first-occurrence argmax
    float bv = score; int bi = a;
    #pragma unroll
    for (int m = 16; m; m >>= 1) {
        float ov = __shfl_xor(bv, m, 32);
        int   oi = __shfl_xor(bi, m, 32);
        if (ov > bv || (ov == bv && oi < bi)) { bv = ov; bi = oi; }
    }
    // softmax over rewards
    float mx = r;
    #pragma unroll
    for (int m = 16; m; m >>= 1) mx = fmaxf(mx, __shfl_xor(mx, m, 32));
    float e = __expf(r - mx);
    float s = wave_sum(e);
    out[O_PROBS + a] = e / s;
    float selr = __shfl(r, bi, 32);
    if (a == 0) {
        out[0]     = selr;
        out[O_ARM] = (float)bi;
        sel[0]     = bi;
    }
}

// ---------- 7: U_new = U + G[sel]^2, streamed per-element ----------
__global__ void unew_kernel(const float* __restrict__ U,   const float* __restrict__ fcW,
                            const float* __restrict__ dg,  const float* __restrict__ xin2,
                            const float* __restrict__ h2,  const float* __restrict__ hn2,
                            const int* __restrict__ selp,  float* __restrict__ out) {
    int s = selp[0];
    unsigned stride = gridDim.x * blockDim.x;
    for (unsigned p = blockIdx.x * blockDim.x + threadIdx.x; p < TOTALP; p += stride) {
        __builtin_prefetch(U + p + stride, 0, 1);   // global_prefetch_b8: next grid-stride chunk
        float g2;
        if (p < N_IH) {
            unsigned r = p / I1, j = p - r * I1;
            float D = fcW[s * HID + (r % HID)] * dg[r];
            g2 = D * D * xin2[j];
        } else if (p < BIH_OFF) {
            unsigned q = p - N_IH, r = q / HID, j = q - r * HID;
            float D = fcW[s * HID + (r % HID)] * dg[r];
            g2 = D * D * h2[j];
        } else if (p < BHH_OFF) {
            unsigned r = p - BIH_OFF;
            float D = fcW[s * HID + (r % HID)] * dg[r];
            g2 = D * D;
        } else if (p < FCW_OFF) {
            unsigned r = p - BHH_OFF;
            float D = fcW[s * HID + (r % HID)] * dg[r];
            g2 = D * D;
        } else if (p < FCB_OFF) {
            unsigned q = p - FCW_OFF, ar = q / HID, j = q - ar * HID;
            g2 = (ar <= (unsigned)s) ? hn2[j] : 0.f;     // cumsum'd fc_W grads
        } else {
            unsigned ar = p - FCB_OFF;
            g2 = (ar <= (unsigned)s) ? 1.f : 0.f;        // cumsum'd fc_b grads
        }
        out[O_U + p] = U[p] + g2;
    }
}

extern "C" void kernel_launch(void* const* d_in, const int* in_sizes, int n_in,
                              void* d_out, int out_size, void* d_ws, size_t ws_size,
                              hipStream_t stream) {
    const float* x    = (const float*)d_in[0];
    const float* prev = (const float*)d_in[1];
    const float* h0   = (const float*)d_in[2];
    const float* c0   = (const float*)d_in[3];
    const float* Wih  = (const float*)d_in[4];
    const float* Whh  = (const float*)d_in[5];
    const float* bih  = (const float*)d_in[6];
    const float* bhh  = (const float*)d_in[7];
    const float* fcW  = (const float*)d_in[8];
    const float* fcb  = (const float*)d_in[9];
    const float* U    = (const float*)d_in[10];
    float* out = (float*)d_out;
    float* ws  = (float*)d_ws;

    float* xin   = ws + 0;       // 513 (pad 544)
    float* xin2  = ws + 544;     // 513
    float* h2    = ws + 1088;    // 768
    float* gates = ws + 1856;    // 3072
    float* dg    = ws + 4928;    // 3072
    float* hn2   = ws + 8000;    // 768
    float* Wr    = ws + 8768;    // 3072
    float* Tfc   = ws + 11840;   // 32
    float* rew   = ws + 11872;   // 32
    float* s2l   = ws + 11904;   // 32
    int*   sel   = (int*)(ws + 11936);

    prep_kernel      <<<1,    1024, 0, stream>>>(x, prev, h0, xin, xin2, h2);
    gates_wmma_kernel<<<192,  32,   0, stream>>>(Wih, Whh, bih, bhh, h0, xin, gates);
    act_kernel       <<<3,    256,  0, stream>>>(gates, c0, out, dg, hn2);
    wr_kernel        <<<384,  256,  0, stream>>>(U, xin2, h2, Wr);
    arm_kernel       <<<32,   32,   0, stream>>>(fcW, fcb, U, out, dg, Wr, hn2, rew, Tfc, s2l);
    select_kernel    <<<1,    32,   0, stream>>>(rew, Tfc, s2l, U, out, sel);
    unew_kernel      <<<4096, 256,  0, stream>>>(U, fcW, dg, xin2, h2, hn2, sel, out);
}